// DifferentiableReconstruction_31662498906489
// MI455X (gfx1250) — compile-verified
//
#include <hip/hip_runtime.h>

typedef __attribute__((ext_vector_type(2))) float v2f;
typedef __attribute__((ext_vector_type(8))) float v8f;

#define VXD 64
#define VVV (64*64*64)     // voxels per batch
#define TT  128            // T
#define NB  2              // B
#define HW  65536          // H*W per slice (C==1)
#define LOG2E 1.4426950408889634f

// ---------------- Pass 1: per-slice intensity mean (bandwidth-bound, 64 MB) ----
__global__ void __launch_bounds__(256)
slice_mean_kernel(const float* __restrict__ slices, float* __restrict__ means) {
    __shared__ float red[256];
    const int slice = blockIdx.x;                 // flat b*T + t
    const float4* p = (const float4*)(slices + (size_t)slice * HW);
    float s = 0.f;
    #pragma unroll 4
    for (int i = threadIdx.x; i < HW / 4; i += 256) {
        float4 q = p[i];                          // global_load_b128, coalesced
        s += (q.x + q.y) + (q.z + q.w);
    }
    red[threadIdx.x] = s;
    __syncthreads();
    for (int off = 128; off > 0; off >>= 1) {
        if (threadIdx.x < off) red[threadIdx.x] += red[threadIdx.x + off];
        __syncthreads();
    }
    if (threadIdx.x == 0) means[slice] = red[0] * (1.0f / (float)HW);
}

// ---------------- Pass 2: WMMA distance tiles + weight accumulation ------------
// dist^2(t,v) = |g_v|^2 + ( |c_t|^2 - 2 g_v . c_t ), computed entirely by WMMA:
//   A (16x4, f32): row t = (-2cx, -2cy, -2cz, |c|^2)   [K0..K3]
//   B (4x16, f32): col v = ( gx ,  gy ,  gz ,  1  )    [K0..K3]
//   C (16x16,f32): broadcast |g_v|^2 per column (constant over M -> same value
//                  in all 8 C VGPRs of a lane, since N = lane&15)
//   D = A x B + C -> 16 t-rows x 16 voxel-cols of dist^2, no VALU fixup needed.
// Weight: exp(1/(dist+1e-6)) == exp2(log2e * rsqrt(d2)) to within 1.2e-6 rel,
// valid because d2 >= 0.75 by construction (integer grid vs half-integer centers).
__global__ void __launch_bounds__(256)
recon_kernel(const float* __restrict__ transforms,
             const int*   __restrict__ sel_idx,
             const float* __restrict__ means,
             float*       __restrict__ out) {
    __shared__ float lds_c[TT * 4];   // per selected slice: -2cx,-2cy,-2cz,|c|^2
    __shared__ float lds_m[TT];       // per selected slice: intensity mean

    const int b   = blockIdx.y;
    const int tid = threadIdx.x;

    if (tid < TT) {
        const int j  = sel_idx[b * TT + tid];                 // gather transform+mean
        const float* tr = transforms + (size_t)(b * TT + j) * 6;
        const float cx = tr[0], cy = tr[1], cz = tr[2];
        lds_c[tid * 4 + 0] = -2.0f * cx;
        lds_c[tid * 4 + 1] = -2.0f * cy;
        lds_c[tid * 4 + 2] = -2.0f * cz;
        lds_c[tid * 4 + 3] = cx * cx + cy * cy + cz * cz;
        lds_m[tid] = means[b * TT + j];
    }
    __syncthreads();

    const int  lane = tid & 31;
    const int  wave = tid >> 5;
    const int  n    = lane & 15;          // voxel column within tile (and A row)
    const bool hi   = lane >= 16;         // half-wave select (K2/K3 vs K0/K1)

    const int   v  = blockIdx.x * 128 + wave * 16 + n;  // flat voxel x*4096+y*64+z
    const float gx = (float)(v >> 12);
    const float gy = (float)((v >> 6) & 63);
    const float gz = (float)(v & 63);
    const float g2 = gx * gx + gy * gy + gz * gz;

    // B matrix fragment (constant across all t-tiles)
    v2f bm;
    bm.x = hi ? gz   : gx;    // VGPR0: K=2 row | K=0 row
    bm.y = hi ? 1.0f : gy;    // VGPR1: K=3 row | K=1 row

    // C matrix fragment: |g|^2 broadcast down the voxel column
    const v8f cacc = {g2, g2, g2, g2, g2, g2, g2, g2};

    float sumw = 0.f, sumwm = 0.f;

    #pragma unroll
    for (int tt = 0; tt < TT / 16; ++tt) {
        const float* cp = &lds_c[(tt * 16 + n) * 4];  // A row M = lane&15
        v2f am;
        am.x = hi ? cp[2] : cp[0];    // VGPR0: K=2 | K=0
        am.y = hi ? cp[3] : cp[1];    // VGPR1: K=3 | K=1

        // (neg_a, A, neg_b, B, c_mod, C, reuse_a, reuse_b) ; D = A*B + C = dist^2
        v8f d = __builtin_amdgcn_wmma_f32_16x16x4_f32(
            false, am, false, bm, (short)0, cacc, false, false);

        const int tofs = tt * 16 + (hi ? 8 : 0);      // C/D: M = r + 8*hi
        #pragma unroll
        for (int r = 0; r < 8; ++r) {
            const float y = __builtin_amdgcn_rsqf(d[r]);        // v_rsq_f32
            const float w = __builtin_amdgcn_exp2f(y * LOG2E);  // v_exp_f32
            sumw  += w;
            sumwm += w * lds_m[tofs + r];             // broadcast LDS read
        }
    }

    // lanes L and L+16 hold disjoint t-halves of the same voxel: fold them
    sumw  += __shfl_xor(sumw,  16, 32);
    sumwm += __shfl_xor(sumwm, 16, 32);

    if (!hi) {
        // single per-voxel normalization; native rcp is fine here too
        out[(size_t)b * VVV + v] = sumwm * __builtin_amdgcn_rcpf(sumw);
    }
}

// ---------------- Host launcher ------------------------------------------------
extern "C" void kernel_launch(void* const* d_in, const int* in_sizes, int n_in,
                              void* d_out, int out_size, void* d_ws, size_t ws_size,
                              hipStream_t stream) {
    const float* slices     = (const float*)d_in[0];  // (B,T,1,H,W) f32
    const float* transforms = (const float*)d_in[1];  // (B,T,6) f32
    const int*   sel_idx    = (const int*)d_in[2];    // (B,T) i32
    float*       out        = (float*)d_out;          // (B,1,64,64,64) f32
    float*       means      = (float*)d_ws;           // B*T floats of scratch

    // Pass 1: 256 slices, one block each, streaming 64 MB total.
    slice_mean_kernel<<<NB * TT, 256, 0, stream>>>(slices, means);

    // Pass 2: each block = 8 waves x 16 voxels = 128 voxels; grid covers V per b.
    dim3 grid(VVV / 128, NB);
    recon_kernel<<<grid, 256, 0, stream>>>(transforms, sel_idx, means, out);
}